// CompactExpandModule_58360015618226
// MI455X (gfx1250) — compile-verified
//
#include <hip/hip_runtime.h>
#include <hip/hip_bf16.h>
#include <stdint.h>

#define B_   8
#define S_   8192
#define D_   1024
#define LC_  2048
#define MAPN 32768   // >= VOCAB (32000), power of two

// ---------------------------------------------------------------------------
// CDNA5 async global<->LDS copy primitives (ASYNCcnt-tracked DMA path).
// Each lane moves 16B: wave32 moves 512B per instruction, no VGPR data regs.
// ---------------------------------------------------------------------------
__device__ __forceinline__ void async_g2l_b128(unsigned lds_byte_off, const void* gsrc) {
    asm volatile("global_load_async_to_lds_b128 %0, %1, off"
                 :
                 : "v"(lds_byte_off), "v"(gsrc)
                 : "memory");
}
__device__ __forceinline__ void async_l2g_b128(void* gdst, unsigned lds_byte_off) {
    asm volatile("global_store_async_from_lds_b128 %0, %1, off"
                 :
                 : "v"(gdst), "v"(lds_byte_off)
                 : "memory");
}
__device__ __forceinline__ void wait_async_zero() {
    asm volatile("s_wait_asynccnt 0" ::: "memory");
}

// ---------------------------------------------------------------------------
// Phase 0: membership byte-map over the vocab
// ---------------------------------------------------------------------------
__global__ void zero_map_kernel(uint32_t* map32) {
    map32[blockIdx.x * 256 + threadIdx.x] = 0u;   // 32 blocks * 256 * 4B = 32KiB
}

__global__ void build_map_kernel(const int* __restrict__ keep, int n_keep,
                                 unsigned char* __restrict__ map) {
    int i = blockIdx.x * 256 + threadIdx.x;
    if (i < n_keep) {
        int id = keep[i];
        if (id >= 0 && id < MAPN) map[id] = 1;
    }
}

// ---------------------------------------------------------------------------
// Phase 1: per-batch stable compaction scan.
// One 1024-thread block per batch; 8 tokens per thread; block-wide
// Hillis-Steele scan in LDS. Emits pos[b][rank], counts[b], sel[b][p].
// ---------------------------------------------------------------------------
__global__ void scan_kernel(const int* __restrict__ token_ids,
                            const unsigned char* __restrict__ map,
                            int* __restrict__ pos,            // [B_][LC_]
                            int* __restrict__ counts,         // [B_]
                            unsigned char* __restrict__ sel)  // [B_][S_]
{
    const int b   = blockIdx.x;
    const int tid = threadIdx.x;          // 0..1023
    const int* tok = token_ids + (size_t)b * S_;

    int m[8];
    int cnt = 0;
    const int base = tid * 8;
#pragma unroll
    for (int j = 0; j < 8; ++j) {
        int t = tok[base + j];
        int k = (t >= 0 && t < MAPN) ? (int)map[t] : 0;
        m[j] = k;
        cnt += k;
    }

    __shared__ int ssum[1024];
    ssum[tid] = cnt;
    __syncthreads();
#pragma unroll
    for (int off = 1; off < 1024; off <<= 1) {
        int v = (tid >= off) ? ssum[tid - off] : 0;
        __syncthreads();
        ssum[tid] += v;
        __syncthreads();
    }
    const int incl  = ssum[tid];
    const int excl  = incl - cnt;
    const int total = ssum[1023];

    int rank = excl;
#pragma unroll
    for (int j = 0; j < 8; ++j) {
        const int p = base + j;
        int selv = 0;
        if (m[j]) {
            if (rank < LC_) {
                pos[b * LC_ + rank] = p;
                selv = 1;
            }
            ++rank;
        }
        sel[(size_t)b * S_ + p] = (unsigned char)selv;
    }
    if (tid == 1023) counts[b] = (total < LC_) ? total : LC_;
}

// ---------------------------------------------------------------------------
// Phase 2a: compacted output. One block per [b, i] row (4KiB).
// Valid row: gather via async global->LDS->global. Invalid row: zeros.
// ---------------------------------------------------------------------------
__global__ void compact_kernel(const float* __restrict__ in,
                               const int* __restrict__ pos,
                               const int* __restrict__ counts,
                               float* __restrict__ outc)
{
    const int i   = blockIdx.x;   // 0..LC_-1
    const int b   = blockIdx.y;   // 0..B_-1
    const int tid = threadIdx.x;  // 0..255

    float* dst = outc + ((size_t)(b * LC_ + i)) * D_ + tid * 4;
    if (i < counts[b]) {                   // block-uniform branch, EXEC all-ones
        const int p = pos[b * LC_ + i];
        const float* src = in + ((size_t)b * S_ + p) * D_ + tid * 4;
        const unsigned lds = (unsigned)(tid * 16);   // dynamic LDS starts at 0
        async_g2l_b128(lds, src);
        wait_async_zero();
        async_l2g_b128(dst, lds);
        wait_async_zero();
    } else {
        *(float4*)dst = make_float4(0.f, 0.f, 0.f, 0.f);
    }
}

// ---------------------------------------------------------------------------
// Phase 2b: expanded output. One block per [b, p] row (4KiB).
// Selected row: async copy from input (likely L2-resident). Else zeros.
// ---------------------------------------------------------------------------
__global__ void expand_kernel(const float* __restrict__ in,
                              const unsigned char* __restrict__ sel,
                              float* __restrict__ oute)
{
    const int p   = blockIdx.x;   // 0..S_-1
    const int b   = blockIdx.y;   // 0..B_-1
    const int tid = threadIdx.x;  // 0..255

    const size_t row = (size_t)b * S_ + p;
    float* dst = oute + row * D_ + tid * 4;
    if (sel[row]) {                        // block-uniform branch
        const float* src = in + row * D_ + tid * 4;
        const unsigned lds = (unsigned)(tid * 16);
        async_g2l_b128(lds, src);
        wait_async_zero();
        async_l2g_b128(dst, lds);
        wait_async_zero();
    } else {
        *(float4*)dst = make_float4(0.f, 0.f, 0.f, 0.f);
    }
}

// ---------------------------------------------------------------------------
// Host launcher
// ---------------------------------------------------------------------------
extern "C" void kernel_launch(void* const* d_in, const int* in_sizes, int n_in,
                              void* d_out, int out_size, void* d_ws, size_t ws_size,
                              hipStream_t stream)
{
    const float* emb  = (const float*)d_in[0];   // [B, S, D] f32
    const int*   tok  = (const int*)d_in[1];     // [B, S] i32
    const int*   keep = (const int*)d_in[2];     // [N_KEEP] i32
    const int n_keep  = in_sizes[2];

    uint8_t* ws = (uint8_t*)d_ws;
    unsigned char* map    = ws;                                   // 32 KiB
    unsigned char* sel    = ws + MAPN;                            // B*S bytes = 64 KiB
    int*           pos    = (int*)(ws + MAPN + (size_t)B_ * S_);  // B*LC*4 = 64 KiB
    int*           counts = (int*)(ws + MAPN + (size_t)B_ * S_ + (size_t)B_ * LC_ * 4);

    float* outc = (float*)d_out;                         // [B, LC, D]
    float* oute = outc + (size_t)B_ * LC_ * D_;          // [B, S, D]

    zero_map_kernel<<<MAPN / (256 * 4), 256, 0, stream>>>((uint32_t*)map);
    build_map_kernel<<<(n_keep + 255) / 256, 256, 0, stream>>>(keep, n_keep, map);
    scan_kernel<<<B_, 1024, 0, stream>>>(tok, map, pos, counts, sel);
    compact_kernel<<<dim3(LC_, B_), 256, 4096, stream>>>(emb, pos, counts, outc);
    expand_kernel<<<dim3(S_, B_), 256, 4096, stream>>>(emb, sel, oute);
}